// Attention_2740189134986
// MI455X (gfx1250) — compile-verified
//
#include <hip/hip_runtime.h>
#include <math.h>

typedef float v2f __attribute__((ext_vector_type(2)));
typedef float v8f __attribute__((ext_vector_type(8)));

#define BATCH 8
#define HEADS 8
#define SEQ   1024
#define DHEAD 64
#define DIM   512
#define INNER (HEADS * DHEAD)   // 512

// ---------------------------------------------------------------------------
// Generic fp32 WMMA GEMM: C = alpha * A @ op(B) (+ bias)
//   A: [M x K] row-major, lda
//   B: TRANS_B ? [N x K] : [K x N], row-major, ldb
//   C: [M x N] row-major, ldc
// Batched over blockIdx.z = bz*H + hz with independent b/h strides.
// Block = 256 threads (8 waves). Block tile 128x64. Wave tile 32x32
// (2x2 of 16x16 f32 WMMA). K tiled by 16 through padded LDS.
// All dims must satisfy M%128==0, N%64==0, K%16==0 (true for every call here).
// ---------------------------------------------------------------------------
template <bool TRANS_B, bool HAS_BIAS>
__global__ __launch_bounds__(256) void wmma_gemm(
    const float* __restrict__ A, int lda, long long sAb, long long sAh,
    const float* __restrict__ B, int ldb, long long sBb, long long sBh,
    float* __restrict__ C, int ldc, long long sCb, long long sCh,
    int M, int N, int K, float alpha, const float* __restrict__ bias, int H) {
  __shared__ float As[128][17];  // +1 pad: conflict-free b64 fragment reads
  __shared__ float Bs[64][17];   // stored transposed: Bs[n][k]

  const int z = blockIdx.z;
  const int bz = z / H, hz = z % H;
  A += bz * sAb + hz * sAh;
  B += bz * sBb + hz * sBh;
  C += bz * sCb + hz * sCh;

  const int m0 = blockIdx.x * 128;
  const int n0 = blockIdx.y * 64;

  const int t = threadIdx.x;
  const int wave = t >> 5;
  const int lane = t & 31;
  const int wm = wave & 3;   // 0..3 -> 32-row strip
  const int wn = wave >> 2;  // 0..1 -> 32-col strip
  const int lhalf = lane >> 4;
  const int l16 = lane & 15;

  v8f acc[2][2] = {};

  for (int k0 = 0; k0 < K; k0 += 16) {
    // ---- stage A tile 128x16 (float4, 64B segments per row) ----
#pragma unroll
    for (int i = 0; i < 2; ++i) {
      const int idx = t + 256 * i;
      const int row = idx >> 2;        // 0..127
      const int kq = (idx & 3) << 2;   // 0,4,8,12
      const float4 val =
          *(const float4*)(A + (long long)(m0 + row) * lda + (k0 + kq));
      As[row][kq + 0] = val.x;
      As[row][kq + 1] = val.y;
      As[row][kq + 2] = val.z;
      As[row][kq + 3] = val.w;
    }
    // ---- stage B tile as Bs[n][k] ----
    if (TRANS_B) {
      // B[N x K]: same pattern as A (64 rows x 4 quads = 256 threads)
      const int row = t >> 2;          // 0..63
      const int kq = (t & 3) << 2;
      const float4 val =
          *(const float4*)(B + (long long)(n0 + row) * ldb + (k0 + kq));
      Bs[row][kq + 0] = val.x;
      Bs[row][kq + 1] = val.y;
      Bs[row][kq + 2] = val.z;
      Bs[row][kq + 3] = val.w;
    } else {
      // B[K x N]: coalesced 256B rows, scatter-transpose into LDS
#pragma unroll
      for (int i = 0; i < 4; ++i) {
        const int idx = t + 256 * i;   // 0..1023
        const int kk = idx >> 6;       // 0..15
        const int n = idx & 63;
        Bs[n][kk] = B[(long long)(k0 + kk) * ldb + (n0 + n)];
      }
    }
    __syncthreads();

    // ---- 4 k-steps of 4: fragments per ISA 16x4/4x16 f32 layout ----
    // lanes 0-15 hold K {+0,+1}, lanes 16-31 hold K {+2,+3}
#pragma unroll
    for (int kk = 0; kk < 4; ++kk) {
      const int kbase = kk * 4 + 2 * lhalf;
      v2f a[2], b[2];
#pragma unroll
      for (int tm = 0; tm < 2; ++tm) {
        const int row = wm * 32 + tm * 16 + l16;
        a[tm][0] = As[row][kbase];
        a[tm][1] = As[row][kbase + 1];
      }
#pragma unroll
      for (int tn = 0; tn < 2; ++tn) {
        const int col = wn * 32 + tn * 16 + l16;
        b[tn][0] = Bs[col][kbase];
        b[tn][1] = Bs[col][kbase + 1];
      }
#pragma unroll
      for (int tm = 0; tm < 2; ++tm)
#pragma unroll
        for (int tn = 0; tn < 2; ++tn)
          acc[tm][tn] = __builtin_amdgcn_wmma_f32_16x16x4_f32(
              false, a[tm], false, b[tn], (short)0, acc[tm][tn], false, false);
    }
    __syncthreads();
  }

  // ---- store: VGPR r -> row M=r (lanes 0-15) / M=r+8 (lanes 16-31) ----
#pragma unroll
  for (int tm = 0; tm < 2; ++tm) {
#pragma unroll
    for (int tn = 0; tn < 2; ++tn) {
      const int col = n0 + wn * 32 + tn * 16 + l16;
      const float bval = HAS_BIAS ? bias[col] : 0.0f;
#pragma unroll
      for (int r = 0; r < 8; ++r) {
        const int row = m0 + wm * 32 + tm * 16 + r + 8 * lhalf;
        float val = alpha * acc[tm][tn][r];
        if (HAS_BIAS) val += bval;
        C[(long long)row * ldc + col] = val;
      }
    }
  }
}

// ---------------------------------------------------------------------------
// Sinkhorn (eps = 1):  u = log_mu - lse_j(v_j - C_ij)
//                      v = log_nu - lse_i(u_i - C_ij)
// ---------------------------------------------------------------------------
#define LOG_N 6.931471805599453f  // log(1024); log_mu = log_nu = -LOG_N

__global__ __launch_bounds__(256) void sinkhorn_row_pass(
    const float* __restrict__ Cmat, const float* __restrict__ v,
    float* __restrict__ u) {
  const int wave = threadIdx.x >> 5;
  const int lane = threadIdx.x & 31;
  const long long rowg = (long long)blockIdx.x * 8 + wave;  // (bh*1024 + i)
  const long long bh = rowg >> 10;
  const float* crow = Cmat + rowg * SEQ;
  const float* vv = v + bh * SEQ;

  float m = -1e30f, s = 0.0f;
  for (int j = lane; j < SEQ; j += 32) {
    const float x = vv[j] - crow[j];
    const float nm = fmaxf(m, x);
    s = s * __expf(m - nm) + __expf(x - nm);
    m = nm;
  }
#pragma unroll
  for (int off = 16; off >= 1; off >>= 1) {
    const float om = __shfl_xor(m, off, 32);
    const float os = __shfl_xor(s, off, 32);
    const float nm = fmaxf(m, om);
    s = s * __expf(m - nm) + os * __expf(om - nm);
    m = nm;
  }
  if (lane == 0) u[rowg] = -LOG_N - (m + __logf(s));
}

__global__ __launch_bounds__(256) void sinkhorn_col_pass(
    const float* __restrict__ Cmat, const float* __restrict__ u,
    float* __restrict__ v) {
  const int bh = blockIdx.z;
  const int j = blockIdx.x * 256 + threadIdx.x;
  const float* cs = Cmat + (long long)bh * SEQ * SEQ;
  const float* uu = u + (long long)bh * SEQ;

  float m = -1e30f, s = 0.0f;
  for (int i = 0; i < SEQ; ++i) {  // coalesced: 256 consecutive floats/row
    const float x = uu[i] - cs[(long long)i * SEQ + j];
    const float nm = fmaxf(m, x);
    s = s * __expf(m - nm) + __expf(x - nm);
    m = nm;
  }
  v[(long long)bh * SEQ + j] = -LOG_N - (m + __logf(s));
}

// attn_ij = exp(u_i + v_j - C_ij) * n, in place (float4 vectorized)
__global__ __launch_bounds__(256) void sinkhorn_transform(
    float* __restrict__ Cmat, const float* __restrict__ u,
    const float* __restrict__ v) {
  const long long idx4 = (long long)blockIdx.x * 256 + threadIdx.x;
  const int j4 = (int)(idx4 & 255);
  const long long rest = idx4 >> 8;
  const int i = (int)(rest & 1023);
  const long long bh = rest >> 10;

  float4* p = (float4*)Cmat + idx4;
  const float4 c = *p;
  const float ui = u[bh * SEQ + i];
  const float* vp = v + bh * SEQ + j4 * 4;
  float4 r;
  r.x = __expf(ui + vp[0] - c.x) * 1024.0f;
  r.y = __expf(ui + vp[1] - c.y) * 1024.0f;
  r.z = __expf(ui + vp[2] - c.z) * 1024.0f;
  r.w = __expf(ui + vp[3] - c.w) * 1024.0f;
  *p = r;
}

__global__ __launch_bounds__(256) void zero_kernel(float* __restrict__ p) {
  p[(long long)blockIdx.x * 256 + threadIdx.x] = 0.0f;
}

// ---------------------------------------------------------------------------
extern "C" void kernel_launch(void* const* d_in, const int* in_sizes, int n_in,
                              void* d_out, int out_size, void* d_ws,
                              size_t ws_size, hipStream_t stream) {
  const float* x = (const float*)d_in[0];      // [8,1024,512]
  const float* W_qkv = (const float*)d_in[1];  // [512,1536]
  const float* W_out = (const float*)d_in[2];  // [512,512]
  const float* b_out = (const float*)d_in[3];  // [512]

  float* outF = (float*)d_out;                      // [8,1024,512]
  float* attn = outF + (size_t)BATCH * SEQ * DIM;   // [8,8,1024,1024] in-place C

  float* qkv = (float*)d_ws;                              // 8*1024*1536
  float* outh = qkv + (size_t)BATCH * SEQ * 3 * INNER;    // 8*1024*512 ([b,n,h*d])
  float* u = outh + (size_t)BATCH * SEQ * INNER;          // 64*1024
  float* v = u + (size_t)BATCH * HEADS * SEQ;             // 64*1024

  const long long sQKVb = (long long)SEQ * 3 * INNER;
  const long long sAttnB = (long long)HEADS * SEQ * SEQ;
  const long long sAttnH = (long long)SEQ * SEQ;

  // 1. qkv = x @ W_qkv                       [8192x512]@[512x1536]
  wmma_gemm<false, false><<<dim3(8192 / 128, 1536 / 64, 1), 256, 0, stream>>>(
      x, DIM, 0, 0, W_qkv, 3 * INNER, 0, 0, qkv, 3 * INNER, 0, 0,
      BATCH * SEQ, 3 * INNER, DIM, 1.0f, nullptr, 1);

  // 2. dots = (q @ k^T) * scale  -> attn region (batched over b,h)
  wmma_gemm<true, false><<<dim3(SEQ / 128, SEQ / 64, BATCH * HEADS), 256, 0,
                           stream>>>(
      qkv, 3 * INNER, sQKVb, DHEAD,               // q: col offset h*64
      qkv + INNER, 3 * INNER, sQKVb, DHEAD,       // k: col offset 512 + h*64
      attn, SEQ, sAttnB, sAttnH,
      SEQ, SEQ, DHEAD, 0.125f, nullptr, HEADS);

  // 3. v0 = 0; 3 Sinkhorn iterations (row pass -> u, col pass -> v)
  zero_kernel<<<(BATCH * HEADS * SEQ) / 256, 256, 0, stream>>>(v);
  for (int it = 0; it < 3; ++it) {
    sinkhorn_row_pass<<<(BATCH * HEADS * SEQ) / 8, 256, 0, stream>>>(attn, v, u);
    sinkhorn_col_pass<<<dim3(SEQ / 256, 1, BATCH * HEADS), 256, 0, stream>>>(
        attn, u, v);
  }

  // 4. attn = exp(u + v - C) * n, in place
  sinkhorn_transform<<<(unsigned)((long long)BATCH * HEADS * SEQ * SEQ / 4 / 256),
                       256, 0, stream>>>(attn, u, v);

  // 5. outh[b, i, h*64+d] = attn @ v_heads   (batched over b,h)
  wmma_gemm<false, false><<<dim3(SEQ / 128, DHEAD / 64, BATCH * HEADS), 256, 0,
                            stream>>>(
      attn, SEQ, sAttnB, sAttnH,
      qkv + 2 * INNER, 3 * INNER, sQKVb, DHEAD,   // v: col 1024 + h*64
      outh, INNER, (long long)SEQ * INNER, DHEAD,
      SEQ, DHEAD, SEQ, 1.0f, nullptr, HEADS);

  // 6. final = outh @ W_out + b_out          [8192x512]@[512x512]
  wmma_gemm<false, true><<<dim3(8192 / 128, DIM / 64, 1), 256, 0, stream>>>(
      outh, INNER, 0, 0, W_out, DIM, 0, 0, outF, DIM, 0, 0,
      BATCH * SEQ, DIM, INNER, 1.0f, b_out, 1);
}